// TokenChoiceTopKRouter_58772332479040
// MI455X (gfx1250) — compile-verified
//
#include <hip/hip_runtime.h>
#include <math.h>

typedef __attribute__((ext_vector_type(2))) float v2f;
typedef __attribute__((ext_vector_type(8))) float v8f;
typedef __attribute__((ext_vector_type(4))) unsigned int u32x4;
typedef __attribute__((ext_vector_type(4))) int i32x4;
typedef __attribute__((ext_vector_type(8))) int i32x8;

#define DIM 2048
#define NTOK 16384
#define NE 64
#define KCHUNK 64
#define XSTR (KCHUNK + 4)   // padded LDS row stride -> bank = 4*row + k (conflict-free b64)
#define WSTR (KCHUNK + 4)
#define REPS 1e-10f

#if __has_include(<hip/amd_detail/amd_gfx1250_TDM.h>)
#define TDM_ARGS6 1
#endif

// Issue one TDM 2-D tile load: 16 rows x KCHUNK f32 from global (row stride DIM)
// into LDS at lds_byte_addr, padding 4 dwords after every 64 dwords so rows land
// at XSTR-float stride. Wave-level op; tracked by TENSORcnt.
__device__ __forceinline__ void tdm_load_x_tile(unsigned lds_byte_addr,
                                                unsigned long long gaddr) {
    u32x4 g0;
    g0[0] = 1u;                                            // count=1 (valid), user mode
    g0[1] = lds_byte_addr;                                 // lds_addr
    g0[2] = (unsigned)(gaddr & 0xffffffffu);               // global_addr[31:0]
    g0[3] = (unsigned)((gaddr >> 32) & 0x1ffffffu)         // global_addr[56:32]
          | (2u << 30);                                    // type=2 ("image")
    i32x8 g1;
    g1[0] = (int)((2u << 16)     // data_size = 4B
                | (1u << 20)     // pad_enable
                | (5u << 22)     // pad_interval: 64 dwords
                | (3u << 25));   // pad_amount: 4 dwords
    g1[1] = (int)(((unsigned)DIM & 0xffffu) << 16);        // tensor_dim0[15:0]
    g1[2] = (int)(((unsigned)NTOK & 0xffffu) << 16);       // dim0 hi=0 | tensor_dim1[15:0]
    g1[3] = (int)(((unsigned)KCHUNK) << 16);               // dim1 hi=0 | tile_dim0
    g1[4] = 16;                                            // tile_dim1=16, tile_dim2=0
    g1[5] = DIM;                                           // tensor_dim0_stride[31:0]
    g1[6] = 0;                                             // stride hi | dim1_stride lo
    g1[7] = 0;
    i32x4 gz = {};
#ifdef TDM_ARGS6
    i32x8 gz8 = {};
    __builtin_amdgcn_tensor_load_to_lds(g0, g1, gz, gz, gz8, 0);
#else
    __builtin_amdgcn_tensor_load_to_lds(g0, g1, gz, gz, 0);
#endif
}

// ---------------------------------------------------------------------------
// Kernel A: fused gate GEMM (fp32 WMMA 16x16x4, TDM-staged A, LDS-transposed B)
// + softmax + clip + top-2 + per-block rz partials. One wave = 16 tok x 64 exp.
// ---------------------------------------------------------------------------
__global__ __launch_bounds__(256) void router_gemm_kernel(
    const float* __restrict__ x, const float* __restrict__ W,
    const float* __restrict__ bias, float* __restrict__ topv,
    int* __restrict__ topi, float* __restrict__ rzpart)
{
    __shared__ float xl[8 * 16 * XSTR];   // per-wave 16 x KCHUNK tiles (padded rows)
    __shared__ float WlT[NE * WSTR];      // W chunk, expert-major (transposed)
    __shared__ float red[8];

    const int tid  = threadIdx.x;
    const int lane = tid & 31;
    const int wave = tid >> 5;
    const int half = lane >> 4;
    const int l15  = lane & 15;
    const int koff = half * 2;            // A/B vgpr0 holds K0 / K2 per half-wave

    const int uw = __builtin_amdgcn_readfirstlane(wave);   // wave-uniform (SGPR)
    const int token_base = blockIdx.x * 128 + uw * 16;
    const unsigned xl_base = (unsigned)__builtin_amdgcn_readfirstlane(
        (int)(unsigned)(uintptr_t)&xl[uw * 16 * XSTR]);
    const unsigned long long x_ga =
        (unsigned long long)(uintptr_t)x +
        ((unsigned long long)token_base * DIM) * 4ull;

    const float* xw = &xl[uw * 16 * XSTR + l15 * XSTR];

    v8f acc[4] = {};   // 4 expert tiles of 16 -> experts 0..63

    for (int kc = 0; kc < DIM; kc += KCHUNK) {
        __syncthreads();                       // previous chunk fully consumed
        tdm_load_x_tile(xl_base, x_ga + (unsigned long long)kc * 4ull);
        // stage W chunk transposed (overlaps with in-flight TDM)
        for (int i = tid; i < KCHUNK * NE; i += 256) {
            int kk = i >> 6;
            int ee = i & 63;
            WlT[ee * WSTR + kk] = W[(size_t)kc * NE + i];
        }
        __builtin_amdgcn_s_wait_tensorcnt(0);  // this wave's A tile resident
        __syncthreads();                       // W chunk resident for all waves

        #pragma unroll
        for (int k = 0; k < KCHUNK; k += 4) {
            v2f a = *(const v2f*)(xw + k + koff);
            #pragma unroll
            for (int t = 0; t < 4; ++t) {
                v2f bf = *(const v2f*)(&WlT[(l15 + 16 * t) * WSTR + k + koff]);
                acc[t] = __builtin_amdgcn_wmma_f32_16x16x4_f32(
                    false, a, false, bf, (short)0, acc[t], false, false);
            }
        }
    }

    // add bias
    #pragma unroll
    for (int t = 0; t < 4; ++t) {
        float be = bias[l15 + 16 * t];
        #pragma unroll
        for (int r = 0; r < 8; ++r) acc[t][r] += be;
    }

    // per-token softmax + clip + rz accumulation + stable top-2
    float rzacc = 0.0f;
    #pragma unroll
    for (int r = 0; r < 8; ++r) {
        float sv[4];
        float m = -INFINITY;
        #pragma unroll
        for (int t = 0; t < 4; ++t) { sv[t] = acc[t][r]; m = fmaxf(m, sv[t]); }
        #pragma unroll
        for (int mask = 1; mask <= 8; mask <<= 1) m = fmaxf(m, __shfl_xor(m, mask, 32));
        float sum = 0.0f;
        #pragma unroll
        for (int t = 0; t < 4; ++t) { sv[t] = expf(sv[t] - m); sum += sv[t]; }
        #pragma unroll
        for (int mask = 1; mask <= 8; mask <<= 1) sum += __shfl_xor(sum, mask, 32);
        float inv = 1.0f / sum;

        float tv1 = -1.0f, tv2 = -1.0f;
        int   ti1 = 0,     ti2 = 0;
        #pragma unroll
        for (int t = 0; t < 4; ++t) {
            float s = sv[t] * inv;
            s = fminf(fmaxf(s, REPS), 1.0f - REPS);
            rzacc += s / (1.0f - s);          // exp(log-odds) for logsumexp
            int e = l15 + 16 * t;
            if (s > tv1)      { tv2 = tv1; ti2 = ti1; tv1 = s; ti1 = e; }
            else if (s > tv2) { tv2 = s;   ti2 = e; }
        }
        // butterfly merge of top-2 across the 16-lane group (ties -> lower idx)
        #pragma unroll
        for (int mask = 1; mask <= 8; mask <<= 1) {
            float w1 = __shfl_xor(tv1, mask, 32);
            int   j1 = __shfl_xor(ti1, mask, 32);
            float w2 = __shfl_xor(tv2, mask, 32);
            int   j2 = __shfl_xor(ti2, mask, 32);
            if (w1 > tv1 || (w1 == tv1 && j1 < ti1)) {
                float nv2; int ni2;
                if (tv1 > w2 || (tv1 == w2 && ti1 < j2)) { nv2 = tv1; ni2 = ti1; }
                else                                     { nv2 = w2;  ni2 = j2;  }
                tv1 = w1; ti1 = j1; tv2 = nv2; ti2 = ni2;
            } else if (w1 > tv2 || (w1 == tv2 && j1 < ti2)) {
                tv2 = w1; ti2 = j1;
            }
        }
        if (l15 == 0) {
            int token = token_base + r + half * 8;
            int slot  = token * 2;
            topv[slot]     = tv1;  topv[slot + 1] = tv2;
            topi[slot]     = ti1;  topi[slot + 1] = ti2;
        }
    }

    // per-block rz partial (fixed order -> deterministic)
    #pragma unroll
    for (int mask = 1; mask <= 16; mask <<= 1) rzacc += __shfl_xor(rzacc, mask, 32);
    if (lane == 0) red[wave] = rzacc;
    __syncthreads();
    if (tid == 0) {
        float s = 0.0f;
        for (int w = 0; w < 8; ++w) s += red[w];
        rzpart[blockIdx.x] = s;
    }
}

// ---------------------------------------------------------------------------
// Kernel B: per-block expert histograms (64 blocks x 512 slots)
// ---------------------------------------------------------------------------
__global__ __launch_bounds__(512) void hist_kernel(
    const int* __restrict__ topi, int* __restrict__ blockHist)
{
    __shared__ int h[NE];
    int tid = threadIdx.x;
    if (tid < NE) h[tid] = 0;
    __syncthreads();
    int e = topi[blockIdx.x * 512 + tid];
    atomicAdd(&h[e], 1);
    __syncthreads();
    if (tid < NE) blockHist[blockIdx.x * NE + tid] = h[tid];
}

// ---------------------------------------------------------------------------
// Kernel C: expert totals, exclusive scan, per-(block,expert) base offsets
// ---------------------------------------------------------------------------
__global__ __launch_bounds__(64) void scan_kernel(
    const int* __restrict__ blockHist, int* __restrict__ blockBase,
    float* __restrict__ out_hist)
{
    __shared__ int totals[NE];
    __shared__ int base[NE];
    int e = threadIdx.x;
    int tot = 0;
    for (int b = 0; b < 64; ++b) tot += blockHist[b * NE + e];
    totals[e] = tot;
    __syncthreads();
    if (e == 0) {
        int run = 0;
        for (int i = 0; i < NE; ++i) { base[i] = run; run += totals[i]; }
    }
    __syncthreads();
    int run = base[e];
    for (int b = 0; b < 64; ++b) { blockBase[b * NE + e] = run; run += blockHist[b * NE + e]; }
    out_hist[e] = (float)totals[e];
}

// ---------------------------------------------------------------------------
// Kernel D: stable scatter (counting sort by expert id, token order kept)
// ---------------------------------------------------------------------------
__global__ __launch_bounds__(512) void scatter_kernel(
    const int* __restrict__ topi, const float* __restrict__ topv,
    const int* __restrict__ blockBase, float* __restrict__ out)
{
    __shared__ int keys[512];
    int tid  = threadIdx.x;
    int slot = blockIdx.x * 512 + tid;
    int e    = topi[slot];
    keys[tid] = e;
    __syncthreads();
    int rank = 0;
    for (int j = 0; j < tid; ++j) rank += (keys[j] == e);
    int pos = blockBase[blockIdx.x * NE + e] + rank;
    out[pos]          = topv[slot];                 // top_scores_sorted
    out[32768 + pos]  = (float)(slot >> 1);         // token_indices (perm // 2)
}

// ---------------------------------------------------------------------------
// Kernel E: rz_loss = mean_b log(sum_blocks partial_b)  (fixed order)
// ---------------------------------------------------------------------------
__global__ void rz_final_kernel(const float* __restrict__ rzpart,
                                float* __restrict__ out)
{
    if (threadIdx.x == 0 && blockIdx.x == 0) {
        float acc = 0.0f;
        for (int bt = 0; bt < 4; ++bt) {
            float s = 0.0f;
            for (int i = 0; i < 32; ++i) s += rzpart[bt * 32 + i];
            acc += logf(s);
        }
        out[65600] = acc * 0.25f;
    }
}

extern "C" void kernel_launch(void* const* d_in, const int* in_sizes, int n_in,
                              void* d_out, int out_size, void* d_ws, size_t ws_size,
                              hipStream_t stream) {
    const float* x    = (const float*)d_in[0];
    const float* W    = (const float*)d_in[1];
    const float* bias = (const float*)d_in[2];
    float* out = (float*)d_out;

    char* ws = (char*)d_ws;
    float* topv      = (float*)(ws + 0);        // 32768 f
    int*   topi      = (int*)  (ws + 131072);   // 32768 i
    int*   blockHist = (int*)  (ws + 262144);   // 64*64 i
    int*   blockBase = (int*)  (ws + 278528);   // 64*64 i
    float* rzpart    = (float*)(ws + 294912);   // 128 f

    router_gemm_kernel<<<128, 256, 0, stream>>>(x, W, bias, topv, topi, rzpart);
    hist_kernel<<<64, 512, 0, stream>>>(topi, blockHist);
    scan_kernel<<<1, 64, 0, stream>>>(blockHist, blockBase, out + 65536);
    scatter_kernel<<<64, 512, 0, stream>>>(topi, topv, blockBase, out);
    rz_final_kernel<<<1, 32, 0, stream>>>(rzpart, out);
}